// Feature_extractor_2D_3D_89043261981395
// MI455X (gfx1250) — compile-verified
//
#include <hip/hip_runtime.h>

typedef __attribute__((ext_vector_type(2))) float v2f;
typedef __attribute__((ext_vector_type(8))) float v8f;

namespace {
constexpr int D0 = 240, H0 = 144, W0 = 240;
constexpr int V  = D0 * H0 * W0;          // 8,294,400 voxels
constexpr int C0 = 12;
constexpr int P  = 480 * 640;             // 307,200 pixels
constexpr int D1 = 120, H1 = 72, W1 = 120;
constexpr int N1 = D1 * H1 * W1;          // 1,036,800
constexpr int D2 = 60,  H2 = 36, W2 = 60;
constexpr int N2 = D2 * H2 * W2;          // 129,600
}

// ---- CDNA5 async global->LDS staging (ASYNCcnt path), caller must __syncthreads() ----
__device__ __forceinline__ void async_stage_to_lds(const float* __restrict__ g,
                                                   float* l, int count,
                                                   int tid, int nthreads) {
  for (int i = tid; i < count; i += nthreads) {
    unsigned       lo = (unsigned)(uintptr_t)(l + i);            // low 32b = LDS offset
    unsigned long long ga = (unsigned long long)(uintptr_t)(g + i);
    asm volatile("global_load_async_to_lds_b32 %0, %1, off"
                 :: "v"(lo), "v"(ga) : "memory");
  }
  asm volatile("s_wait_asynccnt 0x0" ::: "memory");
}

// ---------------- zero fill (float4) ----------------
__global__ void k_zero4(float4* __restrict__ p, int n4) {
  int t = blockIdx.x * blockDim.x + threadIdx.x;
  if (t < n4) p[t] = make_float4(0.f, 0.f, 0.f, 0.f);
}

// ---------------- scatter 2D -> 3D ----------------
__global__ void k_scatter(const float* __restrict__ feat, const int* __restrict__ depth,
                          float* __restrict__ seg) {
  int pIx = blockIdx.x * blockDim.x + threadIdx.x;
  if (pIx >= P) return;
  int idx = depth[pIx];
  if (idx > 0 && idx < V) {
#pragma unroll
    for (int c = 0; c < C0; ++c)
      seg[(size_t)c * V + idx] = feat[(size_t)c * P + pIx];
  }
}

// ---------------- 3^3 avg-pool hole fill ----------------
__global__ void k_pool_fill(const float* __restrict__ seg, float* __restrict__ segf) {
  int t = blockIdx.x * blockDim.x + threadIdx.x;
  if (t >= C0 * V) return;
  int c = t / V, v = t - c * V;
  int x = v % W0;  int r = v / W0;
  int y = r % H0;  int z = r / H0;
  float s = seg[t];
  float out = s;
  if (s == 0.0f) {
    const float* base = seg + (size_t)c * V;
    float sum = 0.f;
#pragma unroll
    for (int dz = -1; dz <= 1; ++dz) {
      int zz = z + dz; if ((unsigned)zz >= (unsigned)D0) continue;
#pragma unroll
      for (int dy = -1; dy <= 1; ++dy) {
        int yy = y + dy; if ((unsigned)yy >= (unsigned)H0) continue;
        const float* row = base + ((size_t)zz * H0 + yy) * W0;
#pragma unroll
        for (int dx = -1; dx <= 1; ++dx) {
          int xx = x + dx; if ((unsigned)xx >= (unsigned)W0) continue;
          sum += row[xx];
        }
      }
    }
    out = sum * (1.0f / 27.0f);
  }
  segf[t] = out;
}

// ---------------- downsample 1: conv(12->4,k3,s2,p1) ++ maxpool2, relu ----------------
__global__ void k_ds1(const float* __restrict__ segf, const float* __restrict__ w,
                      float* __restrict__ x1) {
  __shared__ float lw[4 * 12 * 27];
  async_stage_to_lds(w, lw, 4 * 12 * 27, threadIdx.x, blockDim.x);
  __syncthreads();
  int t = blockIdx.x * blockDim.x + threadIdx.x;
  if (t >= N1) return;
  int ox = t % W1; int r = t / W1;
  int oy = r % H1; int oz = r / H1;
  float acc[4] = {0.f, 0.f, 0.f, 0.f};
  for (int c = 0; c < C0; ++c) {
    const float* xc = segf + (size_t)c * V;
#pragma unroll
    for (int dz = 0; dz < 3; ++dz) {
      int iz = 2 * oz - 1 + dz; if ((unsigned)iz >= (unsigned)D0) continue;
#pragma unroll
      for (int dy = 0; dy < 3; ++dy) {
        int iy = 2 * oy - 1 + dy; if ((unsigned)iy >= (unsigned)H0) continue;
#pragma unroll
        for (int dx = 0; dx < 3; ++dx) {
          int ix = 2 * ox - 1 + dx; if ((unsigned)ix >= (unsigned)W0) continue;
          float vv = xc[((size_t)iz * H0 + iy) * W0 + ix];
          int tap = dz * 9 + dy * 3 + dx;
#pragma unroll
          for (int co = 0; co < 4; ++co) acc[co] += lw[(co * 12 + c) * 27 + tap] * vv;
        }
      }
    }
  }
#pragma unroll
  for (int co = 0; co < 4; ++co)
    x1[(((size_t)co * D1 + oz) * H1 + oy) * W1 + ox] = fmaxf(acc[co], 0.f);
  // maxpool(2,2,2) of the 12 input channels -> output channels 4..15
  for (int c = 0; c < C0; ++c) {
    const float* xc = segf + (size_t)c * V;
    float m = -INFINITY;
#pragma unroll
    for (int dz = 0; dz < 2; ++dz)
#pragma unroll
      for (int dy = 0; dy < 2; ++dy)
#pragma unroll
        for (int dx = 0; dx < 2; ++dx)
          m = fmaxf(m, xc[((size_t)(2 * oz + dz) * H0 + (2 * oy + dy)) * W0 + (2 * ox + dx)]);
    x1[(((size_t)(4 + c) * D1 + oz) * H1 + oy) * W1 + ox] = fmaxf(m, 0.f);
  }
}

// ---------------- generic 1x1x1 conv (+optional residual), relu ----------------
__global__ void k_conv1x1(const float* __restrict__ x, const float* __restrict__ w,
                          const float* __restrict__ res, float* __restrict__ out,
                          int Cin, int Cout, int Nv) {
  int t = blockIdx.x * blockDim.x + threadIdx.x;
  if (t >= Cout * Nv) return;
  int m = t / Nv, v = t - m * Nv;
  float acc = res ? res[t] : 0.f;
  for (int c = 0; c < Cin; ++c)
    acc += w[m * Cin + c] * x[(size_t)c * Nv + v];
  out[t] = fmaxf(acc, 0.f);
}

// ---- generic small bottleneck conv, kernel (kz,ky,kx) "same" pad, +bias +res1 +res2, relu ----
__global__ void k_bconv(const float* __restrict__ x, const float* __restrict__ w,
                        const float* __restrict__ bias,
                        const float* __restrict__ res1, const float* __restrict__ res2,
                        float* __restrict__ out,
                        int C, int Dd, int Hh, int Ww, int kz, int ky, int kx) {
  int Nv = Dd * Hh * Ww;
  int t = blockIdx.x * blockDim.x + threadIdx.x;
  if (t >= C * Nv) return;
  int m = t / Nv, v = t - m * Nv;
  int xx = v % Ww; int r = v / Ww;
  int yy = r % Hh; int zz = r / Hh;
  float acc = bias[m];
  if (res1) acc += res1[t];
  if (res2) acc += res2[t];
  int hz = kz >> 1, hy = ky >> 1, hx = kx >> 1;
  for (int c = 0; c < C; ++c) {
    const float* xc = x + (size_t)c * Nv;
    for (int dz = 0; dz < kz; ++dz) {
      int pz = zz + dz - hz; if ((unsigned)pz >= (unsigned)Dd) continue;
      for (int dy = 0; dy < ky; ++dy) {
        int py = yy + dy - hy; if ((unsigned)py >= (unsigned)Hh) continue;
        for (int dx = 0; dx < kx; ++dx) {
          int px = xx + dx - hx; if ((unsigned)px >= (unsigned)Ww) continue;
          acc += w[(((m * C + c) * kz + dz) * ky + dy) * kx + dx] *
                 xc[((size_t)pz * Hh + py) * Ww + px];
        }
      }
    }
  }
  out[t] = fmaxf(acc, 0.f);
}

// ---------------- downsample 2 conv (16->16,k3,s2,p1) as WMMA implicit GEMM ----------------
// D[16 co x 16 pos] = sum_{k<432} A[co,k] * B[k,pos],  k = cin*27 + tap
__device__ __forceinline__ float ds2_loadB(const float* __restrict__ x,
                                           int k, int oz, int oy, int ox) {
  int cin = k / 27;        int tt = k - cin * 27;
  int dz = tt / 9;         int t2 = tt - dz * 9;
  int dy = t2 / 3;         int dx = t2 - dy * 3;
  int iz = 2 * oz - 1 + dz, iy = 2 * oy - 1 + dy, ix = 2 * ox - 1 + dx;
  if ((unsigned)iz < (unsigned)D1 && (unsigned)iy < (unsigned)H1 && (unsigned)ix < (unsigned)W1)
    return x[(((size_t)cin * D1 + iz) * H1 + iy) * W1 + ix];
  return 0.f;
}

__global__ void __launch_bounds__(128) k_ds2_conv_wmma(const float* __restrict__ x,
                                                       const float* __restrict__ w,
                                                       float* __restrict__ x2) {
  __shared__ float lw[16 * 432];     // A[co][k], contiguous copy of w_ds2
  async_stage_to_lds(w, lw, 16 * 432, threadIdx.x, 128);
  __syncthreads();

  int wid  = blockIdx.x * (blockDim.x >> 5) + (threadIdx.x >> 5);   // wave32
  int lane = threadIdx.x & 31;
  int tx = wid & 3;  int r = wid >> 2;
  int oy = r % H2;   int oz = r / H2;          // oz < 60

  int half = lane >> 4;        // 0: K pair {0,1}; 1: K pair {2,3}  (ISA 16x4 f32 A layout)
  int mk   = lane & 15;        // A row (M) and B column (N) for this lane
  int ox   = tx * 16 + mk;     // output x position for the B/D column
  int kb   = half * 2;

  v8f acc = {};
#pragma unroll 4
  for (int k0 = 0; k0 < 432; k0 += 4) {
    int ka = k0 + kb;
    v2f a, b;
    a.x = lw[mk * 432 + ka];
    a.y = lw[mk * 432 + ka + 1];
    b.x = ds2_loadB(x, ka,     oz, oy, ox);
    b.y = ds2_loadB(x, ka + 1, oz, oy, ox);
    acc = __builtin_amdgcn_wmma_f32_16x16x4_f32(false, a, false, b,
                                                (short)0, acc, false, false);
  }
  // D layout: lanes 0-15 -> M = rr, lanes 16-31 -> M = rr + 8 ; N = lane & 15
#pragma unroll
  for (int rr = 0; rr < 8; ++rr) {
    int m = rr + 8 * half;
    if (ox < W2)
      x2[(((size_t)m * D2 + oz) * H2 + oy) * W2 + ox] = fmaxf(acc[rr], 0.f);
  }
}

// ---------------- downsample 2 maxpool part (channels 16..31) ----------------
__global__ void k_ds2_pool(const float* __restrict__ x1b, float* __restrict__ x2) {
  int t = blockIdx.x * blockDim.x + threadIdx.x;
  if (t >= 16 * N2) return;
  int c = t / N2, v = t - c * N2;
  int ox = v % W2; int r = v / W2;
  int oy = r % H2; int oz = r / H2;
  const float* xc = x1b + (size_t)c * N1;
  float m = -INFINITY;
#pragma unroll
  for (int dz = 0; dz < 2; ++dz)
#pragma unroll
    for (int dy = 0; dy < 2; ++dy)
#pragma unroll
      for (int dx = 0; dx < 2; ++dx)
        m = fmaxf(m, xc[((size_t)(2 * oz + dz) * H1 + (2 * oy + dy)) * W1 + (2 * ox + dx)]);
  x2[(((size_t)(16 + c) * D2 + oz) * H2 + oy) * W2 + ox] = fmaxf(m, 0.f);
}

// ============================================================================
extern "C" void kernel_launch(void* const* d_in, const int* in_sizes, int n_in,
                              void* d_out, int out_size, void* d_ws, size_t ws_size,
                              hipStream_t stream) {
  const float* feat   = (const float*)d_in[0];
  const int*   depth  = (const int*)  d_in[1];
  const float* w_ds1  = (const float*)d_in[2];
  const float* b1_win = (const float*)d_in[3];
  const float* b1_w133= (const float*)d_in[4];  const float* b1_b133 = (const float*)d_in[5];
  const float* b1_w331= (const float*)d_in[6];  const float* b1_b331 = (const float*)d_in[7];
  const float* b1_w313= (const float*)d_in[8];  const float* b1_b313 = (const float*)d_in[9];
  const float* b1_wout= (const float*)d_in[10];
  const float* w_ds2  = (const float*)d_in[11];
  const float* b2_win = (const float*)d_in[12];
  const float* b2_w133= (const float*)d_in[13]; const float* b2_b133 = (const float*)d_in[14];
  const float* b2_w331= (const float*)d_in[15]; const float* b2_b331 = (const float*)d_in[16];
  const float* b2_w313= (const float*)d_in[17]; const float* b2_b313 = (const float*)d_in[18];
  const float* b2_wout= (const float*)d_in[19];
  float* out = (float*)d_out;
  float* ws  = (float*)d_ws;

  // workspace layout (floats); regions recycled once their producers' consumers finish
  float* seg  = ws;                            // 12*V
  float* segf = ws + (size_t)C0 * V;           // 12*V
  float* x1   = seg;                           // 16*N1  (reuses seg region)
  float* y0   = seg  + (size_t)16 * N1;        // 4*N1
  float* y1   = y0   + (size_t)4  * N1;
  float* y2   = y1   + (size_t)4  * N1;
  float* y3   = y2   + (size_t)4  * N1;
  float* x1b  = segf;                          // 16*N1  (reuses segf region)
  float* x2   = seg;                           // 32*N2  (x1/y* dead by then)
  float* z0   = x2   + (size_t)32 * N2;        // 8*N2
  float* z1   = z0   + (size_t)8  * N2;
  float* z2   = z1   + (size_t)8  * N2;
  float* z3   = z2   + (size_t)8  * N2;

  const int BT = 256;

  // 1) zero + scatter into segres
  {
    int n4 = (C0 * V) / 4;
    k_zero4<<<(n4 + BT - 1) / BT, BT, 0, stream>>>((float4*)seg, n4);
    k_scatter<<<(P + BT - 1) / BT, BT, 0, stream>>>(feat, depth, seg);
  }
  // 2) avg-pool hole fill
  k_pool_fill<<<(C0 * V + BT - 1) / BT, BT, 0, stream>>>(seg, segf);
  // 3) downsample 1 (conv ++ pool, relu)
  k_ds1<<<(N1 + BT - 1) / BT, BT, 0, stream>>>(segf, w_ds1, x1);
  // 4) bottleneck 1 (mid = 4, C = 16)
  k_conv1x1<<<(4 * N1 + BT - 1) / BT, BT, 0, stream>>>(x1, b1_win, nullptr, y0, 16, 4, N1);
  k_bconv<<<(4 * N1 + BT - 1) / BT, BT, 0, stream>>>(y0, b1_w133, b1_b133, nullptr, nullptr,
                                                     y1, 4, D1, H1, W1, 1, 3, 3);
  k_bconv<<<(4 * N1 + BT - 1) / BT, BT, 0, stream>>>(y1, b1_w331, b1_b331, y1, nullptr,
                                                     y2, 4, D1, H1, W1, 3, 3, 1);
  k_bconv<<<(4 * N1 + BT - 1) / BT, BT, 0, stream>>>(y2, b1_w313, b1_b313, y2, y1,
                                                     y3, 4, D1, H1, W1, 3, 1, 3);
  k_conv1x1<<<(16 * N1 + BT - 1) / BT, BT, 0, stream>>>(y3, b1_wout, x1, x1b, 4, 16, N1);
  // 5) downsample 2: WMMA conv (ch 0..15) + maxpool (ch 16..31)
  {
    int waves = 4 * H2 * D2;                   // 4 x-tiles of 16 over W2=60
    k_ds2_conv_wmma<<<waves / 4, 128, 0, stream>>>(x1b, w_ds2, x2);
    k_ds2_pool<<<(16 * N2 + BT - 1) / BT, BT, 0, stream>>>(x1b, x2);
  }
  // 6) bottleneck 2 (mid = 8, C = 32), final residual conv writes d_out
  k_conv1x1<<<(8 * N2 + BT - 1) / BT, BT, 0, stream>>>(x2, b2_win, nullptr, z0, 32, 8, N2);
  k_bconv<<<(8 * N2 + BT - 1) / BT, BT, 0, stream>>>(z0, b2_w133, b2_b133, nullptr, nullptr,
                                                     z1, 8, D2, H2, W2, 1, 3, 3);
  k_bconv<<<(8 * N2 + BT - 1) / BT, BT, 0, stream>>>(z1, b2_w331, b2_b331, z1, nullptr,
                                                     z2, 8, D2, H2, W2, 3, 3, 1);
  k_bconv<<<(8 * N2 + BT - 1) / BT, BT, 0, stream>>>(z2, b2_w313, b2_b313, z2, z1,
                                                     z3, 8, D2, H2, W2, 3, 1, 3);
  k_conv1x1<<<(32 * N2 + BT - 1) / BT, BT, 0, stream>>>(z3, b2_wout, x2, out, 8, 32, N2);
}